// ESNBase_12421045420520
// MI455X (gfx1250) — compile-verified
//
#include <hip/hip_runtime.h>
#include <math.h>

typedef __attribute__((ext_vector_type(2))) float v2f;
typedef __attribute__((ext_vector_type(8))) float v8f;

#define SEQ   1024
#define BATCH 64
#define IND   512
#define RES   2048

__device__ __forceinline__ float fast_tanh(float x) {
    // tanh(x) = 1 - 2/(exp(2x)+1); exp overflow/underflow saturates to +/-1.
    float e = __expf(2.0f * x);
    return 1.0f - __fdividef(2.0f, e + 1.0f);
}

// ---------------------------------------------------------------------------
// Kernel 1: xin = x @ W_in^T + b   (M=65536, K=512, N=2048)
// 128-thread block (4 waves), 64(M) x 32(N) tile per block; each wave owns a
// 16x32 strip = two independent 16x16 WMMA accumulator chains sharing the A
// fragment. Constant-trip staging loops => no exec-mask guards.
// ---------------------------------------------------------------------------
__global__ __launch_bounds__(128)
void esn_in_proj(const float* __restrict__ A,   // M x K row-major (x)
                 const float* __restrict__ W,   // N x K row-major (W_in)
                 const float* __restrict__ bias,
                 float* __restrict__ D) {       // M x N (states buffer)
    constexpr int K = IND;
    constexpr int N = RES;
    __shared__ float Al[64][36];   // stride 36: 16B-aligned rows + bank spread
    __shared__ float Bl[32][36];

    const int tid  = threadIdx.x;
    const int wave = tid >> 5;     // 0..3 (wave32)
    const int lane = tid & 31;
    const int lrow = lane & 15;
    const int lhi  = lane >> 4;
    const int m0   = blockIdx.y * 64;
    const int n0   = blockIdx.x * 32;

    v8f c0, c1;
    const float b0 = bias[n0 + lrow];
    const float b1 = bias[n0 + 16 + lrow];
    #pragma unroll
    for (int j = 0; j < 8; ++j) { c0[j] = b0; c1[j] = b1; }

    for (int k0 = 0; k0 < K; k0 += 32) {
        __syncthreads();
        #pragma unroll
        for (int p = 0; p < 4; ++p) {            // A tile 64x32, b128 moves
            int i = tid * 4 + p * 512;
            int r = i >> 5, kk = i & 31;
            *(float4*)&Al[r][kk] =
                *(const float4*)&A[(size_t)(m0 + r) * K + (k0 + kk)];
        }
        #pragma unroll
        for (int p = 0; p < 2; ++p) {            // W tile 32x32
            int i = tid * 4 + p * 512;
            int r = i >> 5, kk = i & 31;
            *(float4*)&Bl[r][kk] =
                *(const float4*)&W[(size_t)(n0 + r) * K + (k0 + kk)];
        }
        __syncthreads();

        #pragma unroll
        for (int kc = 0; kc < 32; kc += 4) {
            v2f af  = *(const v2f*)&Al[16 * wave + lrow][kc + 2 * lhi];
            v2f bf0 = *(const v2f*)&Bl[lrow][kc + 2 * lhi];
            v2f bf1 = *(const v2f*)&Bl[16 + lrow][kc + 2 * lhi];
            c0 = __builtin_amdgcn_wmma_f32_16x16x4_f32(false, af, false, bf0,
                                                       (short)0, c0, false, false);
            c1 = __builtin_amdgcn_wmma_f32_16x16x4_f32(false, af, false, bf1,
                                                       (short)0, c1, false, false);
        }
    }

    #pragma unroll
    for (int j = 0; j < 8; ++j) {
        int row = m0 + 16 * wave + 8 * lhi + j;
        D[(size_t)row * N + (n0 + lrow)]      = c0[j];
        D[(size_t)row * N + (n0 + 16 + lrow)] = c1[j];
    }
}

// ---------------------------------------------------------------------------
// Kernel 2: one recurrence step, in place: U (=xin_t) -> tanh(U + Hprev@W^T)
// K-split across the 4 waves (512 each); wave-private LDS staging => zero
// block barriers in the K loop; 4 independent accumulator chains per wave;
// speculative prefetch of the next K tile. One __syncthreads for the
// cross-wave reduction + tanh. Grid: 128 blocks x 128 threads.
// ---------------------------------------------------------------------------
__global__ __launch_bounds__(128)
void esn_step(const float* __restrict__ Hprev,  // 64 x RES (states[t-1])
              const float* __restrict__ W,      // RES x RES row-major (W_hh)
              float* __restrict__ U,            // in: xin_t, out: h_t
              int has_prev) {
    constexpr int N  = RES;         // 2048
    constexpr int K  = RES;         // 2048
    constexpr int KW = K / 4;       // 512 K-columns per wave

    __shared__ float Al[4][64][36]; // wave-private A (Hprev) staging
    __shared__ float Bl[4][16][36]; // wave-private W staging
    __shared__ float Pl[4][64][16]; // per-wave partial results

    const int tid  = threadIdx.x;
    const int wave = tid >> 5;
    const int lane = tid & 31;
    const int lrow = lane & 15;
    const int lhi  = lane >> 4;
    const int n0   = blockIdx.x * 16;
    const int kw0  = wave * KW;

    if (has_prev) {                 // uniform branch: EXEC all-ones at WMMA
        v8f acc[4];
        #pragma unroll
        for (int rt = 0; rt < 4; ++rt)
            #pragma unroll
            for (int j = 0; j < 8; ++j) acc[rt][j] = 0.f;

        for (int k0 = 0; k0 < KW; k0 += 32) {
            // Wave-private staging (same-wave LDS ops are in-order; the
            // compiler inserts s_wait_dscnt) -- no __syncthreads needed.
            #pragma unroll
            for (int p = 0; p < 16; ++p) {       // A tile 64x32, b128 moves
                int i = lane * 4 + p * 128;
                int r = i >> 5, kk = i & 31;
                *(float4*)&Al[wave][r][kk] =
                    *(const float4*)&Hprev[(size_t)r * K + (kw0 + k0 + kk)];
            }
            #pragma unroll
            for (int p = 0; p < 4; ++p) {        // W tile 16x32
                int i = lane * 4 + p * 128;
                int r = i >> 5, kk = i & 31;
                *(float4*)&Bl[wave][r][kk] =
                    *(const float4*)&W[(size_t)(n0 + r) * K + (kw0 + k0 + kk)];
            }

            // Speculative prefetch of the next K tile (global_prefetch_b8;
            // OOB addresses on the final iteration are silently dropped).
            __builtin_prefetch(&Hprev[(size_t)lane * K + (kw0 + k0 + 32)], 0, 0);
            __builtin_prefetch(&Hprev[(size_t)(lane + 32) * K + (kw0 + k0 + 32)], 0, 0);
            __builtin_prefetch(&W[(size_t)(n0 + lrow) * K + (kw0 + k0 + 32)], 0, 0);

            #pragma unroll
            for (int kc = 0; kc < 32; kc += 4) {
                v2f bf = *(const v2f*)&Bl[wave][lrow][kc + 2 * lhi];
                #pragma unroll
                for (int rt = 0; rt < 4; ++rt) {   // 4 independent chains
                    v2f af = *(const v2f*)&Al[wave][16 * rt + lrow][kc + 2 * lhi];
                    acc[rt] = __builtin_amdgcn_wmma_f32_16x16x4_f32(
                        false, af, false, bf, (short)0, acc[rt], false, false);
                }
            }
        }

        // Dump partials: VGPR j of tile rt holds P[16*rt + 8*lhi + j][lrow]
        #pragma unroll
        for (int rt = 0; rt < 4; ++rt)
            #pragma unroll
            for (int j = 0; j < 8; ++j)
                Pl[wave][16 * rt + 8 * lhi + j][lrow] = acc[rt][j];
    }

    __syncthreads();

    // Reduce 4 K-partials, add xin, tanh, store in place (each element read
    // and written only by this thread; Hprev lives in states[t-1]).
    #pragma unroll
    for (int p = 0; p < 8; ++p) {                // coalesced, no guards
        int i = tid + p * 128;
        int r = i >> 4, n = i & 15;
        size_t idx = (size_t)r * N + (n0 + n);
        float v = U[idx];
        if (has_prev)
            v += Pl[0][r][n] + Pl[1][r][n] + Pl[2][r][n] + Pl[3][r][n];
        U[idx] = fast_tanh(v);
    }
}

// ---------------------------------------------------------------------------
extern "C" void kernel_launch(void* const* d_in, const int* in_sizes, int n_in,
                              void* d_out, int out_size, void* d_ws, size_t ws_size,
                              hipStream_t stream) {
    (void)in_sizes; (void)n_in; (void)out_size; (void)d_ws; (void)ws_size;

    const float* x    = (const float*)d_in[0];  // (SEQ, BATCH, IN)
    const float* W_in = (const float*)d_in[1];  // (RES, IN)
    const float* W_hh = (const float*)d_in[2];  // (RES, RES)
    const float* b    = (const float*)d_in[3];  // (RES,)
    float* out = (float*)d_out;                 // states ++ h_n

    // 1) Input projection GEMM straight into the states buffer.
    dim3 g1(RES / 32, (SEQ * BATCH) / 64);
    esn_in_proj<<<g1, 128, 0, stream>>>(x, W_in, b, out);

    // 2) Sequential recurrence; stream order serializes the 1024 steps.
    dim3 g2(RES / 16, 1);
    for (int t = 0; t < SEQ; ++t) {
        const float* hprev = out + (size_t)(t > 0 ? t - 1 : 0) * BATCH * RES;
        esn_step<<<g2, 128, 0, stream>>>(hprev, W_hh,
                                         out + (size_t)t * BATCH * RES, t > 0);
    }

    // 3) h_n = states[SEQ-1], appended after the states block.
    hipMemcpyAsync(out + (size_t)SEQ * BATCH * RES,
                   out + (size_t)(SEQ - 1) * BATCH * RES,
                   sizeof(float) * BATCH * RES, hipMemcpyDeviceToDevice, stream);
}